// H2I_74895639708134
// MI455X (gfx1250) — compile-verified
//
#include <hip/hip_runtime.h>
#include <cstdint>

#define ROW_LEN 4096
#define RADIUS  64
#define PADDED  (ROW_LEN + RADIUS)   // 4160
#define PAD_VAL (-1000.0f)
#define NEG_INF (-3.0e38f)

// One workgroup per row. 256 threads (8 waves, wave32).
__global__ __launch_bounds__(256) void horizon_scan_kernel(
    const float* __restrict__ hf, float* __restrict__ out) {
  __shared__ __align__(16) float s_h[PADDED];  // row + sentinel tail
  __shared__ float s_P[PADDED];                // per-64-block prefix max of g
  __shared__ float s_S[PADDED];                // per-64-block suffix max of g

  const int tid = threadIdx.x;
  const int row = blockIdx.x;
  const float* rowp = hf + (size_t)row * ROW_LEN;

  // ---- Phase 1: async copy of the 16 KB row into LDS (CDNA5 async path) ----
  // LDS byte offset = low 32 bits of the flat address of the shared array.
  uint32_t lds_base = (uint32_t)(uintptr_t)(&s_h[0]);
#pragma unroll
  for (int c = 0; c < 4; ++c) {
    uint32_t goff = (uint32_t)(c * 4096 + tid * 16);  // bytes; 4 KB per chunk
    uint32_t laddr = lds_base + goff;
    // GVS form: vdst = LDS byte address, vaddr = 32-bit byte offset, saddr = base
    asm volatile("global_load_async_to_lds_b128 %0, %1, %2"
                 :
                 : "v"(laddr), "v"(goff), "s"(rowp)
                 : "memory");
  }
  // Sentinel pad region [4096, 4160): plain LDS stores, disjoint from async dests.
  if (tid < RADIUS) s_h[ROW_LEN + tid] = PAD_VAL;
  asm volatile("s_wait_asynccnt 0" ::: "memory");
  __syncthreads();

  // ---- Phase 2: per-64-block scans of g[k] = h[k] - (k mod 64) ----
  // 65 blocks; even thread of each pair does prefix, odd does suffix.
  {
    const int nblk = PADDED / RADIUS;  // 65
    int b = tid >> 1;
    if (b < nblk) {
      int base = b * RADIUS;
      if ((tid & 1) == 0) {
        float run = NEG_INF;
#pragma unroll 4
        for (int j = 0; j < RADIUS; ++j) {
          run = fmaxf(run, s_h[base + j] - (float)j);
          s_P[base + j] = run;
        }
      } else {
        float run = NEG_INF;
#pragma unroll 4
        for (int j = RADIUS - 1; j >= 0; --j) {
          run = fmaxf(run, s_h[base + j] - (float)j);
          s_S[base + j] = run;
        }
      }
    }
  }
  __syncthreads();

  // ---- Phase 3: window max from two LDS lookups, coalesced b128 stores ----
  float* orow = out + (size_t)row * ROW_LEN;
#pragma unroll
  for (int c = 0; c < 4; ++c) {
    int i0 = c * 1024 + tid * 4;
    float r[4];
#pragma unroll
    for (int k = 0; k < 4; ++k) {
      int i = i0 + k;
      int p = (i + 1) & 63;
      float suff = s_S[i + 1] + (float)(p - 1);
      float pref = (p == 0) ? NEG_INF : s_P[i + RADIUS] + (float)(p - 65);
      float m = fmaxf(suff, pref);
      r[k] = fmaxf(m - s_h[i], 0.0f);
    }
    float4 v = make_float4(r[0], r[1], r[2], r[3]);
    *reinterpret_cast<float4*>(orow + i0) = v;
  }
}

extern "C" void kernel_launch(void* const* d_in, const int* in_sizes, int n_in,
                              void* d_out, int out_size, void* d_ws, size_t ws_size,
                              hipStream_t stream) {
  (void)n_in; (void)out_size; (void)d_ws; (void)ws_size;
  const float* hf = (const float*)d_in[0];
  float* out = (float*)d_out;
  int rows = in_sizes[0] / ROW_LEN;  // 1024
  horizon_scan_kernel<<<rows, 256, 0, stream>>>(hf, out);
}